// THLMaxInverseDist_66374424592540
// MI455X (gfx1250) — compile-verified
//
#include <hip/hip_runtime.h>
#include <hip/hip_bf16.h>
#include <math.h>

// Problem constants (match reference setup_inputs: x = [8192, 256] fp32)
#define N 8192
#define KDIM 256
#define LDSPAD 260          // 256 + 4 floats pad -> conflict-free ds_load_b64 pattern
#define ROWS_PER_BLOCK 128  // 8 waves * 16 rows
#define JSEGS 8             // J range split across blockIdx.y
#define JT_PER_SEG ((N / 16) / JSEGS)   // 64 j-tiles per segment

typedef __attribute__((ext_vector_type(2))) float v2f;
typedef __attribute__((ext_vector_type(8))) float v8f;

// ---------------- helper kernels ----------------

__global__ void __launch_bounds__(256) zero_rowloss_kernel(int* rowloss) {
    int i = blockIdx.x * 256 + threadIdx.x;
    if (i < N) rowloss[i] = 0;   // bits of +0.0f, identity for positive-float max
}

// One block per row: L2-normalize the row.
__global__ void __launch_bounds__(256) rownorm_kernel(const float* __restrict__ x,
                                                      float* __restrict__ xn) {
    __shared__ float red[256];
    int r = blockIdx.x, t = threadIdx.x;
    float v = x[r * KDIM + t];
    red[t] = v * v;
    __syncthreads();
    for (int s = 128; s > 0; s >>= 1) {
        if (t < s) red[t] += red[t + s];
        __syncthreads();
    }
    float nrm = sqrtf(red[0]);
    xn[r * KDIM + t] = v / nrm;
}

// Deterministic two-pass column mean. Pass 1: 32 blocks, each sums 256 rows.
__global__ void __launch_bounds__(256) colpart_kernel(const float* __restrict__ xn,
                                                      float* __restrict__ partial) {
    int b = blockIdx.x, t = threadIdx.x;
    float s = 0.f;
    int rbase = b * 256;
    for (int r = 0; r < 256; ++r) s += xn[(rbase + r) * KDIM + t];
    partial[b * 256 + t] = s;
}

// Pass 2: fixed-order combine of 32 partials -> column mean.
__global__ void __launch_bounds__(256) colmean_kernel(const float* __restrict__ partial,
                                                      float* __restrict__ mean) {
    int t = threadIdx.x;
    float s = 0.f;
    for (int b = 0; b < 32; ++b) s += partial[b * 256 + t];
    mean[t] = s * (1.0f / (float)N);
}

// One block per row: subtract column mean (in place), compute row squared norm.
__global__ void __launch_bounds__(256) center_kernel(float* __restrict__ xn,
                                                     const float* __restrict__ mean,
                                                     float* __restrict__ rown) {
    __shared__ float red[256];
    int r = blockIdx.x, t = threadIdx.x;
    float v = xn[r * KDIM + t] - mean[t];
    xn[r * KDIM + t] = v;
    red[t] = v * v;
    __syncthreads();
    for (int s = 128; s > 0; s >>= 1) {
        if (t < s) red[t] += red[t + s];
        __syncthreads();
    }
    if (t == 0) rown[r] = red[0];
}

// ---------------- main fused GEMM + row-max kernel ----------------
// Block: 256 threads = 8 waves. Wave w owns output rows [blockIdx.x*128 + w*16, +16).
// blockIdx.y selects a J segment of 64 16-column tiles.
// Per wave: A strip (16x256 fp32) held in registers (128 VGPRs; wave32 allows 1024,
// launch_bounds(256,1) tells the allocator low occupancy is acceptable so it does
// NOT spill); B tile staged in padded LDS, shared by all 8 waves; K accumulated
// with v_wmma_f32_16x16x4_f32 (exact fp32, matches the fp32 reference).
__global__ void __launch_bounds__(256, 1)
maxinv_wmma_kernel(const float* __restrict__ xc,
                   const float* __restrict__ rown,
                   int* __restrict__ rowloss) {
    __shared__ float Bsh[16 * LDSPAD];

    const int tid  = threadIdx.x;
    const int lane = tid & 31;
    const int wave = tid >> 5;             // 0..7
    const int half = lane >> 4;            // K-half select (0/1)
    const int r16  = lane & 15;            // row (A) / col (B) within tile

    const int iTile = blockIdx.x * ROWS_PER_BLOCK + wave * 16;

    // ---- Load this wave's A strip (16 rows x 256 cols) into registers.
    // Lane layout for V_WMMA_F32_16X16X4_F32 A (16x4): lanes 0-15 -> K={k,k+1},
    // lanes 16-31 -> K={k+2,k+3}, row = lane%16.
    const float* arow = xc + (size_t)(iTile + r16) * KDIM;
    v2f A[64];
#pragma unroll
    for (int t = 0; t < 64; ++t)
        A[t] = *(const v2f*)(arow + t * 4 + 2 * half);

    // Row squared-norms for the 8 output rows this lane contributes to (C layout:
    // VGPR v, lane-half h -> row v + 8*h).
    float ni[8];
#pragma unroll
    for (int v = 0; v < 8; ++v) ni[v] = rown[iTile + v + 8 * half];

    float rmax[8];
#pragma unroll
    for (int v = 0; v < 8; ++v) rmax[v] = 0.f;

    const int jBase = blockIdx.y * JT_PER_SEG;
    const int srow  = tid >> 4;            // B-stage row for this thread
    const int scol  = (tid & 15) * 16;     // B-stage col base for this thread

    for (int jt = 0; jt < JT_PER_SEG; ++jt) {
        const int J = (jBase + jt) * 16;

        __syncthreads();   // previous iteration's LDS readers done
        {
            // Cooperative B stage: 16 rows x 256 floats, 16 floats per thread.
            const float4* src = (const float4*)(xc + (size_t)(J + srow) * KDIM + scol);
            float4* dst = (float4*)(&Bsh[srow * LDSPAD + scol]);
            dst[0] = src[0];
            dst[1] = src[1];
            dst[2] = src[2];
            dst[3] = src[3];
        }
        // Prefetch next J tile toward this WGP while we compute (global_prefetch_b8).
        if (jt + 1 < JT_PER_SEG) {
            __builtin_prefetch(xc + (size_t)(J + 16 + srow) * KDIM + scol, 0, 3);
        }
        __syncthreads();

        // ---- Gram tile: C = A(16x256) * B(16x256)^T via 64 K=4 fp32 WMMAs.
        v8f C = {};
        const float* bb = &Bsh[r16 * LDSPAD + 2 * half];
#pragma unroll
        for (int t = 0; t < 64; ++t) {
            v2f b = *(const v2f*)(bb + t * 4);   // conflict-free: pad 4 -> bank step 4
            C = __builtin_amdgcn_wmma_f32_16x16x4_f32(
                    false, A[t], false, b, (short)0, C, false, false);
        }

        // ---- Epilogue: res = n_i + n_j - 2*G; val = 1/dist (diag modified);
        // butterfly row-max within each 16-lane half.
        const float nj = rown[J + r16];
#pragma unroll
        for (int v = 0; v < 8; ++v) {
            int grow = iTile + v + 8 * half;
            int gcol = J + r16;
            float res = fmaxf(ni[v] + nj - 2.0f * C[v], 1e-30f);
            float val;
            if (grow == gcol) {
                float d = sqrtf(res);            // ~1e-15 after clamp
                val = 1.0f / (10.0f - 9.0f * d); // d + 10*(1-d)
            } else {
                val = 1.0f / sqrtf(res);
            }
            val = fmaxf(val, __shfl_xor(val, 1, 32));
            val = fmaxf(val, __shfl_xor(val, 2, 32));
            val = fmaxf(val, __shfl_xor(val, 4, 32));
            val = fmaxf(val, __shfl_xor(val, 8, 32));
            rmax[v] = fmaxf(rmax[v], val);
        }
    }

    // Combine across J segments. Positive floats order like their int bits.
    if (r16 == 0) {
#pragma unroll
        for (int v = 0; v < 8; ++v)
            atomicMax(&rowloss[iTile + v + 8 * half], __float_as_int(rmax[v]));
    }
}

// Final mean over per-row losses (fixed-order tree -> deterministic).
__global__ void __launch_bounds__(256) mean_kernel(const int* __restrict__ rowloss,
                                                   float* __restrict__ out) {
    __shared__ float red[256];
    int t = threadIdx.x;
    float s = 0.f;
    for (int r = t; r < N; r += 256) s += __int_as_float(rowloss[r]);
    red[t] = s;
    __syncthreads();
    for (int k = 128; k > 0; k >>= 1) {
        if (t < k) red[t] += red[t + k];
        __syncthreads();
    }
    if (t == 0) out[0] = red[0] * (1.0f / (float)N);
}

// ---------------- launcher ----------------

extern "C" void kernel_launch(void* const* d_in, const int* in_sizes, int n_in,
                              void* d_out, int out_size, void* d_ws, size_t ws_size,
                              hipStream_t stream) {
    (void)in_sizes; (void)n_in; (void)out_size; (void)ws_size;

    const float* x = (const float*)d_in[0];
    float* out = (float*)d_out;

    char* ws = (char*)d_ws;
    float* xn      = (float*)ws;                                 // N*K floats (8 MB)
    float* mean    = (float*)(ws + (size_t)N * KDIM * 4);        // 256
    float* rown    = mean + 256;                                 // N
    int*   rowloss = (int*)(rown + N);                           // N
    float* partial = (float*)(rowloss + N);                      // 32*256

    zero_rowloss_kernel<<<(N + 255) / 256, 256, 0, stream>>>(rowloss);
    rownorm_kernel<<<N, 256, 0, stream>>>(x, xn);
    colpart_kernel<<<32, 256, 0, stream>>>(xn, partial);
    colmean_kernel<<<1, 256, 0, stream>>>(partial, mean);
    center_kernel<<<N, 256, 0, stream>>>(xn, mean, rown);

    dim3 grid(N / ROWS_PER_BLOCK, JSEGS);   // 64 x 8 = 512 workgroups
    maxinv_wmma_kernel<<<grid, 256, 0, stream>>>(xn, rown, rowloss);

    mean_kernel<<<1, 256, 0, stream>>>(rowloss, out);
}